// TimeDomainModel_20469814133345
// MI455X (gfx1250) — compile-verified
//
#include <hip/hip_runtime.h>
#include <math.h>

// Balloon-Windkessel BOLD model, 1000 explicit-Euler steps, ALPHA == 1.
//
// With alpha = 1, fout = v, so v and q obey constant-coefficient linear
// recurrences v' = a v + b f, q' = a q + b g (a = 1 - dt/mtt, b = dt/mtt,
// g = f*Ef(f)/E0). We chunk n into 16-step tiles: the chunk-local response is
// the lower-triangular Toeplitz matrix L[r][c] = b*a^(r-c) applied to the
// drive chunk -> computed for all 63 chunks at once as a 16x16 by 16x64 f32
// matmul using V_WMMA_F32_16X16X4_F32 (4 chained K=4 steps per tile, one
// 16x16 tile per wave, 8 waves). Only the tiny 2-state (s,f) recurrence and a
// 63-element carry scan remain serial; the transcendental Ef is elementwise
// and runs fully parallel.

typedef __attribute__((ext_vector_type(2))) float v2f;
typedef __attribute__((ext_vector_type(8))) float v8f;

#define N_STEPS 1000
#define DT      0.01f
#define V0C     0.02f
#define NCHUNK  64            // 63 live chunks + zero padding
#define NPAD    (NCHUNK * 16) // 1024

__launch_bounds__(256)
__global__ void bold_balloon_kernel(const float* __restrict__ p_mtt,
                                    const float* __restrict__ p_sigma,
                                    const float* __restrict__ p_mu,
                                    const float* __restrict__ p_lamb,
                                    const float* __restrict__ p_c,
                                    const float* __restrict__ p_beta,
                                    float* __restrict__ out)
{
    __shared__ float fbuf[NPAD];    // f_n (flow entering step n)
    __shared__ float gbuf[NPAD];    // g_n = f_n * Ef(f_n) / E0
    __shared__ float ubuf[NPAD];    // (L f)[chunk][row]  -> chunk-local v response
    __shared__ float wbuf[NPAD];    // (L g)[chunk][row]  -> chunk-local q response
    __shared__ float Lmat[16][17];  // Toeplitz operator, padded row stride
    __shared__ float apow[17];      // a^0 .. a^16
    __shared__ float carryV[64];    // v at each chunk start
    __shared__ float carryQ[64];    // q at each chunk start

    const int tid = threadIdx.x;

    const float mtt    = p_mtt[0];
    const float sigma  = p_sigma[0];
    const float mu     = p_mu[0];
    const float E0     = p_lamb[0];
    const float cdrive = p_c[0];
    const float beta   = p_beta[0];

    const float bcoef = DT / mtt;         // b
    const float acoef = 1.0f - bcoef;     // a

    // ---- Phase 1: serial 2-state linear recurrence (s, f). Chain = 2 FMAs/step.
    if (tid == 0) {
        float s = 0.0f, f = 1.0f;
        for (int n = 0; n < N_STEPS; ++n) {
            fbuf[n] = f;
            float ds = cdrive - sigma * s - mu * (f - 1.0f);
            float fn = f + DT * s;     // df = s (old s)
            s = s + DT * ds;
            f = fn;
        }
        float ap = 1.0f;
        for (int i = 0; i <= 16; ++i) { apow[i] = ap; ap *= acoef; }
    } else if (tid >= 256 - (NPAD - N_STEPS)) {
        // threads 232..255 zero the 24 padding slots 1000..1023
        fbuf[N_STEPS + (tid - (256 - (NPAD - N_STEPS)))] = 0.0f;
    }
    __syncthreads();

    // ---- Phase 2: elementwise transcendental g_n (parallel) + build L.
    {
        const float lg    = __log2f(1.0f - E0);   // log2(1-E0) < 0
        const float invE0 = 1.0f / E0;
        for (int n = tid; n < NPAD; n += 256) {
            float fv = fbuf[n];
            // 1 - (1-E0)^(1/f)  via native v_exp_f32
            float Ef = 1.0f - __builtin_amdgcn_exp2f(lg / fv);
            gbuf[n] = (n < N_STEPS) ? (fv * Ef * invE0) : 0.0f;
        }
        int r = tid >> 4, c = tid & 15;           // 256 threads = 256 entries
        Lmat[r][c] = (c <= r) ? (bcoef * apow[r - c]) : 0.0f;
    }
    __syncthreads();

    // ---- Phase 3: U = L*F, W = L*G via WMMA f32 16x16x4 (K chained 4x).
    // waves 0-3 -> U over chunk-column tiles 0..3; waves 4-7 -> W.
    {
        const int wave  = tid >> 5;     // 0..7 (wave-uniform)
        const int lane  = tid & 31;
        const int half  = lane >> 4;    // 0 or 1
        const int l15   = lane & 15;
        const float* src = (wave < 4) ? fbuf : gbuf;
        float*       dst = (wave < 4) ? ubuf : wbuf;
        const int ntile = wave & 3;
        const int j     = ntile * 16 + l15;   // chunk (output column) index

        v8f acc = {};
        for (int kk = 0; kk < 4; ++kk) {
            // A (16x4 slice of L): lanes 0-15 hold K=4kk..4kk+1, lanes 16-31 K=4kk+2..3
            // B (4x16 slice of F): mirrored split, column = lane&15
            int k0 = 4 * kk + 2 * half;
            v2f av, bv;
            av.x = Lmat[l15][k0];
            av.y = Lmat[l15][k0 + 1];
            bv.x = src[16 * j + k0];
            bv.y = src[16 * j + k0 + 1];
            acc = __builtin_amdgcn_wmma_f32_16x16x4_f32(
                /*neg_a=*/false, av, /*neg_b=*/false, bv,
                /*c_mod=*/(short)0, acc, /*reuse_a=*/false, /*reuse_b=*/false);
        }
        // D layout: VGPR p holds row m = p + 8*half, column = lane&15
        for (int p = 0; p < 8; ++p) {
            dst[16 * j + p + 8 * half] = acc[p];
        }
    }
    __syncthreads();

    // ---- Phase 4: 63-step carry scan across chunks (ratio a^16). Trivial.
    if (tid == 0) {
        const float a16 = apow[16];
        float cv = 1.0f, cq = 1.0f;               // v_0 = q_0 = 1
        for (int J = 0; J < 63; ++J) {
            carryV[J] = cv; carryQ[J] = cq;
            cv = a16 * cv + ubuf[16 * J + 15];
            cq = a16 * cq + wbuf[16 * J + 15];
        }
    }
    __syncthreads();

    // ---- Phase 5: BOLD output, elementwise.
    // v_{i+1} = a^{r+1} * carryV[I] + (L f_I)[r]   with i = 16I + r
    {
        const float k1  = 7.0f * E0;
        const float k2  = 2.0f;
        const float k3  = 2.0f * E0 - 0.2f;
        const float bv0 = beta * V0C;
        for (int i = tid; i < N_STEPS; i += 256) {
            int I = i >> 4, r = i & 15;
            float ar = apow[r + 1];
            float V  = ar * carryV[I] + ubuf[i];
            float Q  = ar * carryQ[I] + wbuf[i];
            out[i] = bv0 * (k1 * (1.0f - Q) + k2 * (1.0f - Q / V) + k3 * (1.0f - V));
        }
    }
}

extern "C" void kernel_launch(void* const* d_in, const int* in_sizes, int n_in,
                              void* d_out, int out_size, void* d_ws, size_t ws_size,
                              hipStream_t stream) {
    (void)in_sizes; (void)n_in; (void)d_ws; (void)ws_size; (void)out_size;
    const float* mtt   = (const float*)d_in[0];
    const float* sigma = (const float*)d_in[1];
    const float* mu    = (const float*)d_in[2];
    const float* lamb  = (const float*)d_in[3];
    const float* c     = (const float*)d_in[4];
    const float* beta  = (const float*)d_in[5];
    float* out = (float*)d_out;
    bold_balloon_kernel<<<dim3(1), dim3(256), 0, stream>>>(mtt, sigma, mu, lamb, c, beta, out);
}